// GCN_43937515438539
// MI455X (gfx1250) — compile-verified
//
#include <hip/hip_runtime.h>
#include <cstdint>
#include <cstddef>

#define IN_F  128
#define HID_F 64   // HID == OUT == 64

typedef float v2f __attribute__((ext_vector_type(2)));
typedef float v8f __attribute__((ext_vector_type(8)));

// ---------------- degree / normalization ----------------

__global__ void gcn_init_deg(float* deg, int n) {
    int i = blockIdx.x * blockDim.x + threadIdx.x;
    if (i < n) deg[i] = 1.0f;                 // self-loop contributes 1
}

__global__ void gcn_count_deg(const long long* __restrict__ dst,
                              float* __restrict__ deg, int e) {
    int i = blockIdx.x * blockDim.x + threadIdx.x;
    if (i < e) atomicAdd(&deg[(int)dst[i]], 1.0f);
}

__global__ void gcn_make_dinv(float* d, int n) {
    int i = blockIdx.x * blockDim.x + threadIdx.x;
    if (i < n) {
        float v = d[i];
        d[i] = (v > 0.0f) ? rsqrtf(v) : 0.0f;
    }
}

// ---------------- fp32 WMMA GEMM: C[n x 64] = A[n x K] @ B[K x 64] ----------------
// grid.x = ceil(n/16); block = (32, 4): each wave owns a 16x16 output tile.
// A-frag (16x4 f32): lane l -> row M = l%16; VGPR0 = K(kh), VGPR1 = K(kh+1), kh = (l/16)*2.
// B-frag (4x16 f32): lane l -> col N = l%16; same K mapping.
// C/D (16x16 f32, 8 VGPRs): VGPR r -> row r + (l/16)*8, col l%16.

__global__ void gcn_gemm_wmma_f32(const float* __restrict__ A,
                                  const float* __restrict__ B,
                                  float* __restrict__ C,
                                  int nrows, int K) {
    const int rowBase = blockIdx.x * 16;
    const int colBase = threadIdx.y * 16;     // 4 col tiles cover 64 cols
    const int lane    = threadIdx.x;          // wave32
    const int m       = lane & 15;
    const int kh      = (lane >> 4) * 2;      // 0 or 2

    // clamp row for tail tiles (loads stay in bounds; full EXEC preserved for WMMA)
    int row = rowBase + m;
    if (row > nrows - 1) row = nrows - 1;
    const float* arow = A + (size_t)row * K;

    v8f acc = {};
    for (int k = 0; k < K; k += 4) {
        v2f a, b;
        a.x = arow[k + kh];
        a.y = arow[k + kh + 1];
        b.x = B[(size_t)(k + kh)     * 64 + colBase + m];
        b.y = B[(size_t)(k + kh + 1) * 64 + colBase + m];
        acc = __builtin_amdgcn_wmma_f32_16x16x4_f32(
                  /*neg_a=*/false, a, /*neg_b=*/false, b,
                  /*c_mod=*/(short)0, acc, /*reuse_a=*/false, /*reuse_b=*/false);
    }

    const int mrow = (lane >> 4) * 8;
#pragma unroll
    for (int r = 0; r < 8; ++r) {
        int orow = rowBase + mrow + r;
        if (orow < nrows)
            C[(size_t)orow * 64 + colBase + m] = acc[r];
    }
}

// ---------------- edge scatter-add: agg[dst] += h[src] * dinv[s]*dinv[d] ----------------
// thread = (edge, feature j); 64 consecutive threads share one edge -> coalesced.

__global__ void gcn_edge_aggregate(const long long* __restrict__ src,
                                   const long long* __restrict__ dst,
                                   const float* __restrict__ h,
                                   const float* __restrict__ dinv,
                                   float* __restrict__ agg, int e) {
    long long idx = (long long)blockIdx.x * blockDim.x + threadIdx.x;
    long long ei  = idx >> 6;
    int j         = (int)(idx & 63);
    if (ei >= e) return;
    int s = (int)src[ei];
    int d = (int)dst[ei];
    float w = dinv[s] * dinv[d];
    atomicAdd(&agg[(size_t)d * 64 + j], h[(size_t)s * 64 + j] * w);
}

// ---------------- layer-1 epilogue: self-loop + bias + ReLU ----------------

__global__ void gcn_bias_relu_self(const float* __restrict__ agg,
                                   const float* __restrict__ h,
                                   const float* __restrict__ dinv,
                                   const float* __restrict__ bias,
                                   float* __restrict__ act, int n) {
    long long i = (long long)blockIdx.x * blockDim.x + threadIdx.x;
    if (i >= (long long)n * 64) return;
    int node = (int)(i >> 6);
    int j    = (int)(i & 63);
    float di = dinv[node];
    float v  = agg[i] + h[i] * (di * di) + bias[j];
    act[i] = v > 0.0f ? v : 0.0f;
}

// ---------------- layer-2 epilogue: self-loop + bias + row log_softmax ----------------
// one wave per row of 64; lane handles cols (lane, lane+32); shfl reductions (wave32).

__global__ void gcn_finalize_logsoftmax(const float* __restrict__ agg,
                                        const float* __restrict__ h,
                                        const float* __restrict__ dinv,
                                        const float* __restrict__ bias,
                                        float* __restrict__ out, int n) {
    int gtid = blockIdx.x * blockDim.x + threadIdx.x;
    int row  = gtid >> 5;          // wave id -> row (uniform per wave)
    int lane = gtid & 31;
    if (row >= n) return;

    float di = dinv[row];
    float sl = di * di;
    size_t base = (size_t)row * 64;

    float x0 = agg[base + lane]      + h[base + lane]      * sl + bias[lane];
    float x1 = agg[base + lane + 32] + h[base + lane + 32] * sl + bias[lane + 32];

    float mx = fmaxf(x0, x1);
#pragma unroll
    for (int off = 16; off > 0; off >>= 1)
        mx = fmaxf(mx, __shfl_xor(mx, off, 32));

    float s = expf(x0 - mx) + expf(x1 - mx);
#pragma unroll
    for (int off = 16; off > 0; off >>= 1)
        s += __shfl_xor(s, off, 32);

    float lse = mx + logf(s);
    out[base + lane]      = x0 - lse;
    out[base + lane + 32] = x1 - lse;
}

// ---------------- launcher ----------------

extern "C" void kernel_launch(void* const* d_in, const int* in_sizes, int n_in,
                              void* d_out, int out_size, void* d_ws, size_t ws_size,
                              hipStream_t stream) {
    const float*     X   = (const float*)d_in[0];        // [n,128]
    const long long* ei  = (const long long*)d_in[1];    // [2,E] int64
    /* d_in[2] = batch_size (== n), unused */
    const float*     W1  = (const float*)d_in[3];        // [128,64]
    const float*     b1  = (const float*)d_in[4];        // [64]
    const float*     W2  = (const float*)d_in[5];        // [64,64]
    const float*     b2  = (const float*)d_in[6];        // [64]
    float*           out = (float*)d_out;                // [n,64]

    const int n = in_sizes[0] / IN_F;
    const int e = in_sizes[1] / 2;
    const long long* src = ei;
    const long long* dst = ei + e;

    // workspace layout (floats): dinv[nPad] | h[n*64] | agg[n*64] | act[n*64]
    float* ws   = (float*)d_ws;
    size_t nPad = ((size_t)n + 63) & ~(size_t)63;
    float* dinv = ws;
    float* h    = dinv + nPad;
    float* agg  = h    + (size_t)n * 64;
    float* act  = agg  + (size_t)n * 64;

    const int T = 256;
    const long long nf   = (long long)n * 64;
    const long long ef   = (long long)e * 64;
    const int nBlksN     = (n + T - 1) / T;
    const int nBlksE     = (e + T - 1) / T;
    const int nBlksNF    = (int)((nf + T - 1) / T);
    const int nBlksEF    = (int)((ef + T - 1) / T);
    const int nBlksTiles = (n + 15) / 16;
    const int nBlksRows  = (int)(((long long)n * 32 + T - 1) / T);
    const dim3 gemmBlk(32, 4);

    // normalization: deg = 1 + in-degree; dinv = rsqrt(deg)
    gcn_init_deg<<<nBlksN, T, 0, stream>>>(dinv, n);
    hipMemsetAsync(agg, 0, (size_t)n * 64 * sizeof(float), stream);
    gcn_count_deg<<<nBlksE, T, 0, stream>>>(dst, dinv, e);
    gcn_make_dinv<<<nBlksN, T, 0, stream>>>(dinv, n);

    // layer 1: h = X @ W1 ; aggregate ; relu(agg + self + b1)
    gcn_gemm_wmma_f32<<<nBlksTiles, gemmBlk, 0, stream>>>(X, W1, h, n, IN_F);
    gcn_edge_aggregate<<<nBlksEF, T, 0, stream>>>(src, dst, h, dinv, agg, e);
    gcn_bias_relu_self<<<nBlksNF, T, 0, stream>>>(agg, h, dinv, b1, act, n);

    // layer 2: h = act @ W2 ; aggregate ; log_softmax(agg + self + b2)
    hipMemsetAsync(agg, 0, (size_t)n * 64 * sizeof(float), stream);
    gcn_gemm_wmma_f32<<<nBlksTiles, gemmBlk, 0, stream>>>(act, W2, h, n, HID_F);
    gcn_edge_aggregate<<<nBlksEF, T, 0, stream>>>(src, dst, h, dinv, agg, e);
    gcn_finalize_logsoftmax<<<nBlksRows, T, 0, stream>>>(agg, h, dinv, b2, out, n);
}